// LevelWisePool_55482387530048
// MI455X (gfx1250) — compile-verified
//
#include <hip/hip_runtime.h>

// LevelWisePool: segment mean+max over (batch, depth) composite segments.
//   h:          [N, 128] f32
//   node_depth: [N] int (0..255)
//   batch_index:[N] int (0..31)
//   out:        [32, 256, 256] f32 -> out[b, c, l], c<128: mean, c>=128: max
//
// Pipeline: histogram -> scan -> scatter (counting sort by seg), then one
// wave32 per segment reduces its rows. On CDNA5 the row-gather is done by the
// Tensor Data Mover in GATHER MODE (32-bit row indices, 8 rows / 4KB per op),
// double-buffered in LDS with s_wait_tensorcnt for DMA/compute overlap.
// Row indices for each chunk are fetched with one scalar (SMEM) load.

#define NSEG 8192
#define LDIM 256
#define HDIM 128

#if __has_builtin(__builtin_amdgcn_tensor_load_to_lds) && \
    __has_builtin(__builtin_amdgcn_s_wait_tensorcnt)
#define USE_TDM 1
#else
#define USE_TDM 0
#endif

#if USE_TDM
typedef unsigned int v4u __attribute__((ext_vector_type(4)));
typedef int v4i __attribute__((ext_vector_type(4)));
typedef int v8i __attribute__((ext_vector_type(8)));

// Issue one TDM gather of `n` (<=8) rows of 128 f32 from h, selected by
// 32-bit row indices i0..i7, into LDS at byte offset lds_off. Only the first
// n index slots are consumed by the DMA (tile_dim1 = n).
__device__ __forceinline__ void tdm_gather8(const float* hbase,
                                            unsigned int Nrows,
                                            unsigned int lds_off,
                                            unsigned int n,
                                            int i0, int i1, int i2, int i3,
                                            int i4, int i5, int i6, int i7) {
    unsigned long long ga = (unsigned long long)(__UINTPTR_TYPE__)hbase;
    // D# group 0: count=1, gather_mode=1, gather_index_size=1 (32-bit),
    // lds_addr, global_addr[56:0], type=2.
    v4u g0 = { 0xC0000001u,
               lds_off,
               (unsigned int)ga,
               (unsigned int)(ga >> 32) | (2u << 30) };
    // D# group 1: wg_mask=0, data_size=2 (4B); tensor_dim0=128 (bits 79:48);
    // tensor_dim1=Nrows (bits 111:80); tile_dim0=128 (bits 127:112);
    // tile_dim1=n valid indices (bits 143:128); tensor_dim0_stride=128
    // (bits 207:160); dim1 stride ignored in gather mode.
    v8i g1 = { (int)(2u << 16),
               (int)(128u << 16),
               (int)((Nrows & 0xFFFFu) << 16),
               (int)((Nrows >> 16) | (128u << 16)),
               (int)n,
               128,
               0, 0 };
    v4i g2 = { i0, i1, i2, i3 };   // row_index_0..3 (32-bit mode)
    v4i g3 = { i4, i5, i6, i7 };   // row_index_4..7
#if defined(__clang_major__) && (__clang_major__ >= 23)
    v8i gx = { 0, 0, 0, 0, 0, 0, 0, 0 };
    __builtin_amdgcn_tensor_load_to_lds(g0, g1, g2, g3, gx, 0);
#else
    __builtin_amdgcn_tensor_load_to_lds(g0, g1, g2, g3, 0);
#endif
}

// k and n are wave-uniform; fetch 8 indices with one scalar load (order is
// padded by 8 entries, and slots >= n are ignored by the DMA).
__device__ __forceinline__ void tdm_fetch_chunk(const unsigned int* order,
                                                unsigned int k, unsigned int n,
                                                const float* h,
                                                unsigned int Nrows,
                                                unsigned int lds_off) {
    unsigned int ks = (unsigned int)__builtin_amdgcn_readfirstlane((int)k);
    unsigned int ns = (unsigned int)__builtin_amdgcn_readfirstlane((int)n);
    const unsigned int* op = order + ks;
    tdm_gather8(h, Nrows, lds_off, ns,
                (int)op[0], (int)op[1], (int)op[2], (int)op[3],
                (int)op[4], (int)op[5], (int)op[6], (int)op[7]);
}
#endif  // USE_TDM

// One wave32 per segment. 4 waves per 128-thread block.
__global__ void lwp_reduce(const float* __restrict__ h,
                           const unsigned int* __restrict__ offsets,
                           const unsigned int* __restrict__ order,
                           float* __restrict__ out, unsigned int Nrows) {
    int wid = threadIdx.x >> 5;
    int lane = threadIdx.x & 31;
    int wave = blockIdx.x * 4 + wid;

    unsigned int s = offsets[wave];
    unsigned int e = offsets[wave + 1];

    float s0 = 0.f, s1 = 0.f, s2 = 0.f, s3 = 0.f;
    float m0 = -__builtin_inff(), m1 = -__builtin_inff();
    float m2 = -__builtin_inff(), m3 = -__builtin_inff();

#if USE_TDM
    // Two 4KB LDS staging tiles per wave (8 rows x 512B each).
    __shared__ float lds_stage[4 * 2 * 1024];
    unsigned int lds_base = (unsigned int)(__UINTPTR_TYPE__)(&lds_stage[0]);
    unsigned int my_off0 = lds_base + (unsigned int)(wid * 2) * 4096u;

    if (s < e) {
        unsigned int k = s;
        unsigned int ncur = e - k; if (ncur > 8u) ncur = 8u;
        int cur = 0;
        tdm_fetch_chunk(order, k, ncur, h, Nrows, my_off0);
        unsigned int knext = k + ncur;
        while (k < e) {
            unsigned int nnext = 0;
            if (knext < e) {
                nnext = e - knext; if (nnext > 8u) nnext = 8u;
                tdm_fetch_chunk(order, knext, nnext, h, Nrows,
                                my_off0 + (unsigned int)((cur ^ 1) * 4096));
                __builtin_amdgcn_s_wait_tensorcnt(1);  // chunk `cur` done
            } else {
                __builtin_amdgcn_s_wait_tensorcnt(0);
            }
            const float4* buf =
                (const float4*)(lds_stage + (wid * 2 + cur) * 1024);
            for (unsigned int r = 0; r < ncur; ++r) {
                float4 v = buf[r * 32 + lane];
                s0 += v.x; s1 += v.y; s2 += v.z; s3 += v.w;
                m0 = fmaxf(m0, v.x); m1 = fmaxf(m1, v.y);
                m2 = fmaxf(m2, v.z); m3 = fmaxf(m3, v.w);
            }
            k = knext; knext += nnext; ncur = nnext; cur ^= 1;
        }
    }
#else
    const float4* __restrict__ h4 = (const float4*)h;
    for (unsigned int base = s; base < e; base += 32u) {
        unsigned int navail = e - base;
        if (navail > 32u) navail = 32u;
        unsigned int lidx = (unsigned int)lane < navail ? (unsigned int)lane
                                                        : navail - 1u;
        unsigned int myord = order[base + lidx];
        for (unsigned int j = 0; j < navail; ++j) {
            unsigned int idx = (unsigned int)__shfl((int)myord, (int)j, 32);
            if (j + 4u < navail) {
                unsigned int pidx =
                    (unsigned int)__shfl((int)myord, (int)(j + 4u), 32);
                __builtin_prefetch(&h4[(size_t)pidx * 32 + lane], 0, 3);
            }
            float4 v = h4[(size_t)idx * 32 + lane];
            s0 += v.x; s1 += v.y; s2 += v.z; s3 += v.w;
            m0 = fmaxf(m0, v.x); m1 = fmaxf(m1, v.y);
            m2 = fmaxf(m2, v.z); m3 = fmaxf(m3, v.w);
        }
    }
#endif

    unsigned int cntv = e - s;
    int b = wave >> 8;   // seg / 256
    int l = wave & 255;  // seg % 256
    float* __restrict__ obase = out + (size_t)b * (2 * HDIM * LDIM) + l;
    int f = lane * 4;
    if (cntv > 0u) {
        float c = (float)cntv;
        obase[(size_t)(f + 0) * LDIM] = s0 / c;
        obase[(size_t)(f + 1) * LDIM] = s1 / c;
        obase[(size_t)(f + 2) * LDIM] = s2 / c;
        obase[(size_t)(f + 3) * LDIM] = s3 / c;
        obase[(size_t)(HDIM + f + 0) * LDIM] = m0;
        obase[(size_t)(HDIM + f + 1) * LDIM] = m1;
        obase[(size_t)(HDIM + f + 2) * LDIM] = m2;
        obase[(size_t)(HDIM + f + 3) * LDIM] = m3;
    } else {
        obase[(size_t)(f + 0) * LDIM] = 0.f;
        obase[(size_t)(f + 1) * LDIM] = 0.f;
        obase[(size_t)(f + 2) * LDIM] = 0.f;
        obase[(size_t)(f + 3) * LDIM] = 0.f;
        obase[(size_t)(HDIM + f + 0) * LDIM] = 0.f;
        obase[(size_t)(HDIM + f + 1) * LDIM] = 0.f;
        obase[(size_t)(HDIM + f + 2) * LDIM] = 0.f;
        obase[(size_t)(HDIM + f + 3) * LDIM] = 0.f;
    }
}

__global__ void lwp_init(unsigned int* __restrict__ cnt) {
    int i = blockIdx.x * blockDim.x + threadIdx.x;
    if (i < NSEG) cnt[i] = 0u;
}

__global__ void lwp_hist(const int* __restrict__ depth,
                         const int* __restrict__ batch,
                         unsigned int* __restrict__ cnt, int N) {
    int stride = gridDim.x * blockDim.x;
    for (int n = blockIdx.x * blockDim.x + threadIdx.x; n < N; n += stride) {
        int seg = batch[n] * LDIM + depth[n];
        atomicAdd(&cnt[seg], 1u);
    }
}

// Single 1024-thread block: exclusive scan of 8192 counts.
__global__ void lwp_scan(const unsigned int* __restrict__ cnt,
                         unsigned int* __restrict__ offsets,
                         unsigned int* __restrict__ cursor) {
    __shared__ unsigned int sdata[1024];
    int t = threadIdx.x;
    unsigned int vals[8];
    unsigned int lsum = 0u;
#pragma unroll
    for (int j = 0; j < 8; ++j) { vals[j] = cnt[t * 8 + j]; lsum += vals[j]; }
    sdata[t] = lsum;
    __syncthreads();
    for (int off = 1; off < 1024; off <<= 1) {
        unsigned int x = (t >= off) ? sdata[t - off] : 0u;
        __syncthreads();
        sdata[t] += x;
        __syncthreads();
    }
    unsigned int run = sdata[t] - lsum;
#pragma unroll
    for (int j = 0; j < 8; ++j) {
        offsets[t * 8 + j] = run;
        cursor[t * 8 + j] = run;
        run += vals[j];
    }
    if (t == 1023) offsets[NSEG] = sdata[1023];
}

__global__ void lwp_scatter(const int* __restrict__ depth,
                            const int* __restrict__ batch,
                            unsigned int* __restrict__ cursor,
                            unsigned int* __restrict__ order, int N) {
    int stride = gridDim.x * blockDim.x;
    for (int n = blockIdx.x * blockDim.x + threadIdx.x; n < N; n += stride) {
        int seg = batch[n] * LDIM + depth[n];
        unsigned int pos = atomicAdd(&cursor[seg], 1u);
        order[pos] = (unsigned int)n;
    }
}

extern "C" void kernel_launch(void* const* d_in, const int* in_sizes, int n_in,
                              void* d_out, int out_size, void* d_ws, size_t ws_size,
                              hipStream_t stream) {
    const float* h     = (const float*)d_in[0];
    const int*   depth = (const int*)d_in[1];
    const int*   batch = (const int*)d_in[2];
    float*       out   = (float*)d_out;
    int N = in_sizes[1];

    // Workspace layout (u32 units):
    //   [0,      8192)      cnt
    //   [8192,   16385)     offsets (8193)
    //   [16400,  24592)     cursor
    //   [24592,  24592+N+8) order (+8 pad so chunked scalar index reads of 8
    //                       never run past the allocation; padded slots are
    //                       never consumed by the DMA since tile_dim1 = n)
    unsigned int* ws      = (unsigned int*)d_ws;
    unsigned int* cnt     = ws;
    unsigned int* offsets = ws + 8192;
    unsigned int* cursor  = ws + 16400;
    unsigned int* order   = ws + 24592;

    lwp_init<<<(NSEG + 255) / 256, 256, 0, stream>>>(cnt);
    lwp_hist<<<2048, 256, 0, stream>>>(depth, batch, cnt, N);
    lwp_scan<<<1, 1024, 0, stream>>>(cnt, offsets, cursor);
    lwp_scatter<<<2048, 256, 0, stream>>>(depth, batch, cursor, order, N);
    // 8192 segments, 4 waves per 128-thread block -> 2048 blocks.
    lwp_reduce<<<NSEG / 4, 128, 0, stream>>>(h, offsets, order, out,
                                             (unsigned int)N);
}